// HRPLoss_41437844471945
// MI455X (gfx1250) — compile-verified
//
#include <hip/hip_runtime.h>

#define NUM_KPT 17
#define BATCH   64
#define HW      6912            // 96*72
#define HW4     1728            // HW/4
#define BK      (BATCH * NUM_KPT) // 1088

typedef __attribute__((ext_vector_type(2))) float v2f;
typedef __attribute__((ext_vector_type(4))) float f4;
typedef __attribute__((ext_vector_type(8))) float v8f;

// ---------------------------------------------------------------------------
// Wave32 sum-reduction done entirely with V_WMMA_F32_16X16X4_F32 (exact f32,
// no cross-lane shuffle needed):
//  WMMA #1: A[m][0]=p[m], A[m][2]=p[m+16], B=ones -> D1[m][n]=p[m]+p[m+16].
//           Lane n<16 holds rows 0..7 in its 8 D VGPRs, lane n>=16 rows 8..15;
//           per-lane horizontal add t = S_low (lanes 0-15) / S_high (16-31).
//  WMMA #2: same trick on t -> D2[m][n] = t[m]+t[m+16] = S_low+S_high = total,
//           identical in every lane / every D component.
// EXEC must be all-ones: call only from convergent full-wave code.
// ---------------------------------------------------------------------------
__device__ __forceinline__ float wave_reduce_wmma(float p) {
  v2f ones; ones[0] = 1.0f; ones[1] = 1.0f;
  v8f c = {};
  v2f a1; a1[0] = p; a1[1] = 0.0f;
  v8f d1 = __builtin_amdgcn_wmma_f32_16x16x4_f32(
      false, a1, false, ones, (short)0, c, false, false);
  float t = ((d1[0] + d1[1]) + (d1[2] + d1[3])) +
            ((d1[4] + d1[5]) + (d1[6] + d1[7]));
  v2f a2; a2[0] = t; a2[1] = 0.0f;
  v8f d2 = __builtin_amdgcn_wmma_f32_16x16x4_f32(
      false, a2, false, ones, (short)0, c, false, false);
  return d2[0]; // grand total, valid in every lane
}

// Block-wide sum. sdata must have 32 floats. Result valid in wave 0 (incl.
// thread 0). Safe to call repeatedly (barrier at entry).
__device__ __forceinline__ float block_reduce(float v, float* sdata) {
  __syncthreads();
  const int lane = threadIdx.x & 31;
  const int wid  = threadIdx.x >> 5;
  const int nw   = blockDim.x >> 5;
  float w = wave_reduce_wmma(v);
  if (lane == 0) sdata[wid] = w;                     // wave partials
  if (threadIdx.x >= nw && threadIdx.x < 32) sdata[threadIdx.x] = 0.0f;
  __syncthreads();
  float total = 0.0f;
  if (wid == 0) {
    // all 32 lanes of wave 0 execute convergently -> EXEC all-ones for WMMA
    total = wave_reduce_wmma(sdata[lane]);
  }
  return total;
}

__device__ __forceinline__ float hsum4(f4 v) {
  return (v[0] + v[1]) + (v[2] + v[3]);
}

// ---------------------------------------------------------------------------
// Pass 1: one block per (b,k). Streams out1_1/out1_2/out2_1 (single-use, NT
// hint -> don't pollute L2) and targets1/targets2 (reused in pass 2 ->
// regular temporal, stays resident in the 192MB L2).
//   l11_part[bk] = sum (out1_1 - t1)^2
//   l12_part[bk] = sum (o12x*t1 - t2x)^2 + (o12y*t1 - t2y)^2
//   S[bk]        = sum (out2_1 - t1)^2
// ---------------------------------------------------------------------------
__global__ void pass1_kernel(const float* __restrict__ out1_1,
                             const float* __restrict__ out1_2,
                             const float* __restrict__ out2_1,
                             const float* __restrict__ targets1,
                             const float* __restrict__ targets2,
                             float* __restrict__ S,
                             float* __restrict__ l11_part,
                             float* __restrict__ l12_part) {
  const int bk = blockIdx.x;
  const int b  = bk / NUM_KPT;
  const int k  = bk - b * NUM_KPT;
  const size_t base1 = (size_t)bk * HW;
  const size_t basex = ((size_t)b * (2 * NUM_KPT) + k) * HW;
  const size_t basey = basex + (size_t)NUM_KPT * HW;

  const f4* o11  = (const f4*)(out1_1   + base1);
  const f4* t1   = (const f4*)(targets1 + base1);
  const f4* o21  = (const f4*)(out2_1   + base1);
  const f4* o12x = (const f4*)(out1_2   + basex);
  const f4* o12y = (const f4*)(out1_2   + basey);
  const f4* t2x  = (const f4*)(targets2 + basex);
  const f4* t2y  = (const f4*)(targets2 + basey);

  f4 s11v = {0.f, 0.f, 0.f, 0.f};
  f4 s12v = {0.f, 0.f, 0.f, 0.f};
  f4 s21v = {0.f, 0.f, 0.f, 0.f};

  for (int i = threadIdx.x; i < HW4; i += blockDim.x) {
    f4 t  = t1[i];
    f4 lx = t2x[i];
    f4 ly = t2y[i];
    f4 a  = __builtin_nontemporal_load(&o11[i]);
    f4 cc = __builtin_nontemporal_load(&o21[i]);
    f4 px = __builtin_nontemporal_load(&o12x[i]);
    f4 py = __builtin_nontemporal_load(&o12y[i]);

    f4 d1 = a - t;            s11v += d1 * d1;
    f4 dx = px * t - lx;
    f4 dy = py * t - ly;      s12v += dx * dx + dy * dy;
    f4 d2 = cc - t;           s21v += d2 * d2;
  }

  __shared__ float sdata[32];
  float r11 = block_reduce(hsum4(s11v), sdata);
  float r12 = block_reduce(hsum4(s12v), sdata);
  float r21 = block_reduce(hsum4(s21v), sdata);
  if (threadIdx.x == 0) {
    l11_part[bk] = r11;
    l12_part[bk] = r12;
    S[bk]        = r21;
  }
}

// ---------------------------------------------------------------------------
// Top-8-of-17 per batch row via rank counting (matches jax top_k stable
// descending tie-break: j beats k if vj>vk or (vj==vk && j<k)).
// Also produces loss2_1 scalar. 64 threads, one per batch element.
// ---------------------------------------------------------------------------
__global__ void topk_kernel(const float* __restrict__ S,
                            float* __restrict__ mask,
                            float* __restrict__ l21_out) {
  __shared__ float ssum[BATCH];
  const int b = threadIdx.x;

  float vals[NUM_KPT];
#pragma unroll
  for (int k = 0; k < NUM_KPT; ++k) vals[k] = S[b * NUM_KPT + k];

  float ssel = 0.0f;
#pragma unroll
  for (int k = 0; k < NUM_KPT; ++k) {
    const float vk = vals[k];
    int rank = 0;
#pragma unroll
    for (int j = 0; j < NUM_KPT; ++j) {
      const float vj = vals[j];
      if (vj > vk || (vj == vk && j < k)) ++rank;
    }
    const float m = (rank < (NUM_KPT / 2)) ? 1.0f : 0.0f;
    mask[b * NUM_KPT + k] = m;
    ssel += m * vk;
  }

  ssum[b] = ssel;
  __syncthreads();
  if (b == 0) {
    float tot = 0.0f;
    for (int i = 0; i < BATCH; ++i) tot += ssum[i];
    // loss2_1 = sum_selected(S)/(2B) averaged over B*K entries
    *l21_out = tot * (1.0f / (2.0f * (float)BATCH))
                   * (1.0f / (float)(BATCH * NUM_KPT));
  }
}

// ---------------------------------------------------------------------------
// Pass 2: masked loss2_2. m = t1 when mask==1, so the squared diff is
// (t1*(o22 - t2))^2. Blocks with mask==0 write 0 and exit (uniform branch);
// only 8/17 of the blocks stream data; targets hit L2 from pass 1.
// ---------------------------------------------------------------------------
__global__ void pass2_kernel(const float* __restrict__ out2_2,
                             const float* __restrict__ targets1,
                             const float* __restrict__ targets2,
                             const float* __restrict__ mask,
                             float* __restrict__ l22_part) {
  const int bk = blockIdx.x;
  if (mask[bk] == 0.0f) {
    if (threadIdx.x == 0) l22_part[bk] = 0.0f;
    return;
  }
  const int b = bk / NUM_KPT;
  const int k = bk - b * NUM_KPT;
  const size_t base1 = (size_t)bk * HW;
  const size_t basex = ((size_t)b * (2 * NUM_KPT) + k) * HW;
  const size_t basey = basex + (size_t)NUM_KPT * HW;

  const f4* t1   = (const f4*)(targets1 + base1);
  const f4* o22x = (const f4*)(out2_2   + basex);
  const f4* o22y = (const f4*)(out2_2   + basey);
  const f4* t2x  = (const f4*)(targets2 + basex);
  const f4* t2y  = (const f4*)(targets2 + basey);

  f4 acc = {0.f, 0.f, 0.f, 0.f};
  for (int i = threadIdx.x; i < HW4; i += blockDim.x) {
    f4 t  = t1[i];
    f4 lx = t2x[i];
    f4 ly = t2y[i];
    f4 ox = __builtin_nontemporal_load(&o22x[i]);
    f4 oy = __builtin_nontemporal_load(&o22y[i]);
    f4 dx = t * (ox - lx);
    f4 dy = t * (oy - ly);
    acc += dx * dx + dy * dy;
  }

  __shared__ float sdata[32];
  float r = block_reduce(hsum4(acc), sdata);
  if (threadIdx.x == 0) l22_part[bk] = r;
}

// ---------------------------------------------------------------------------
// Finalize: reduce the 1088-entry partial arrays and combine.
// ---------------------------------------------------------------------------
__global__ void finalize_kernel(const float* __restrict__ l11_part,
                                const float* __restrict__ l12_part,
                                const float* __restrict__ l22_part,
                                const float* __restrict__ l21_scalar,
                                const float* __restrict__ weights,
                                float* __restrict__ out) {
  float a = 0.f, bsum = 0.f, c = 0.f;
  for (int i = threadIdx.x; i < BK; i += blockDim.x) {
    a    += l11_part[i];
    bsum += l12_part[i];
    c    += l22_part[i];
  }
  __shared__ float sdata[32];
  float L11 = block_reduce(a,    sdata);
  float L12 = block_reduce(bsum, sdata);
  float L22 = block_reduce(c,    sdata);
  if (threadIdx.x == 0) {
    const float invN = 1.0f / (float)(BATCH * NUM_KPT * HW); // 1/7520256
    const float loss11 = L11 * invN;
    const float loss12 = L12 * invN;
    const float loss22 = L22 * invN;
    const float loss21 = *l21_scalar;
    out[0] = (loss11 + loss21) * weights[0]
           + (loss12 + 5.0f * loss22) * weights[1];
  }
}

// ---------------------------------------------------------------------------
// Launch. Workspace layout (floats):
//   [0,1088)      S
//   [1088,2176)   mask
//   [2176,3264)   l11 partials
//   [3264,4352)   l12 partials
//   [4352,5440)   l22 partials
//   [5440]        loss2_1 scalar
// Every slot is overwritten each call -> no init/memset needed, no atomics.
// ---------------------------------------------------------------------------
extern "C" void kernel_launch(void* const* d_in, const int* in_sizes, int n_in,
                              void* d_out, int out_size, void* d_ws, size_t ws_size,
                              hipStream_t stream) {
  const float* out1_1   = (const float*)d_in[0];
  const float* out1_2   = (const float*)d_in[1];
  const float* out2_1   = (const float*)d_in[2];
  const float* out2_2   = (const float*)d_in[3];
  const float* targets1 = (const float*)d_in[4];
  const float* targets2 = (const float*)d_in[5];
  const float* weights  = (const float*)d_in[6];

  float* ws   = (float*)d_ws;
  float* S    = ws;
  float* mask = ws + BK;
  float* p11  = ws + 2 * BK;
  float* p12  = ws + 3 * BK;
  float* p22  = ws + 4 * BK;
  float* l21  = ws + 5 * BK;

  pass1_kernel<<<BK, 256, 0, stream>>>(out1_1, out1_2, out2_1,
                                       targets1, targets2, S, p11, p12);
  topk_kernel<<<1, BATCH, 0, stream>>>(S, mask, l21);
  pass2_kernel<<<BK, 256, 0, stream>>>(out2_2, targets1, targets2, mask, p22);
  finalize_kernel<<<1, 256, 0, stream>>>(p11, p12, p22, l21, weights,
                                         (float*)d_out);
}